// RNNGNNLayer_5265629904971
// MI455X (gfx1250) — compile-verified
//
#include <hip/hip_runtime.h>

// out[n] = sum_t w[t] * adj[t, n]   (closed form of the (1-lam)/lam EWMA fold)
//   w[0]    = (1-lam)^15
//   w[t>=1] = lam * (1-lam)^(15-t)
//
// Memory-bound: 272 MiB traffic -> ~12us floor at 23.3 TB/s. The T-contraction
// runs on the matrix pipe: V_WMMA_F32_16X16X4_F32 chained over kbase=0,4,8,12.
// NO private arrays anywhere: InstCombine turns `cond ? a[i] : a[j]` into a
// dynamically-indexed array access, which defeats SROA and lowers into huge
// v_cmp/v_cndmask select trees (seen in rounds 1-2). All weights/operands/
// accumulators are named SSA values; lane-dependence costs one v_cndmask each.

typedef float v2f __attribute__((ext_vector_type(2)));
typedef float v8f __attribute__((ext_vector_type(8)));

#define TSTEPS 16

__global__ __launch_bounds__(256, 1) void ewma_fold_wmma(
    const float* __restrict__ adj, const float* __restrict__ lam_p,
    float* __restrict__ out, int nn_i) {
  const size_t NN = (size_t)nn_i;  // elements per time slice (2048*2048)

  // ---- Weights as named scalars (uniform; ~20 VALU ops total) ----
  const float lam = lam_p[0];
  const float oml = 1.0f - lam;
  const float w15 = lam;
  const float w14 = w15 * oml;
  const float w13 = w14 * oml;
  const float w12 = w13 * oml;
  const float w11 = w12 * oml;
  const float w10 = w11 * oml;
  const float w9  = w10 * oml;
  const float w8  = w9 * oml;
  const float w7  = w8 * oml;
  const float w6  = w7 * oml;
  const float w5  = w6 * oml;
  const float w4  = w5 * oml;
  const float w3  = w4 * oml;
  const float w2  = w3 * oml;
  const float w1  = w2 * oml;          // lam * oml^14
  const float o2  = oml * oml;
  const float o4  = o2 * o2;
  const float o8  = o4 * o4;
  const float w0  = o8 * o4 * o2 * oml;  // oml^15 (no division; lam==0 safe)

  const int lane = (int)(threadIdx.x & 31u);
  const int wave = (int)(threadIdx.x >> 5);
  const bool hi = (lane >> 4) != 0;  // lane-half select for WMMA K striping
  const int col = lane & 15;

  // A operands (16x4 f32): VGPR0 = K=(kbase+2*hi), VGPR1 = K+1. All rows
  // identical (broadcast weights). One v_cndmask per component, on named SSA.
  v2f a0, a1, a2, a3;
  a0.x = hi ? w2  : w0;   a0.y = hi ? w3  : w1;
  a1.x = hi ? w6  : w4;   a1.y = hi ? w7  : w5;
  a2.x = hi ? w10 : w8;   a2.y = hi ? w11 : w9;
  a3.x = hi ? w14 : w12;  a3.y = hi ? w15 : w13;

  // One wave computes 64 consecutive outputs (4 WMMA column groups of 16).
  const size_t base = (size_t)blockIdx.x * 512 + (size_t)wave * 64;
  if (base >= NN) return;  // wave-uniform

  if (base + 64 <= NN) {  // wave-uniform: EXEC all-1s on the WMMA path
    // Lane-dependent part of the B address, computed once.
    const size_t lane_off = (hi ? 2 * NN : (size_t)0) + (size_t)col;
    const float* wbase = adj + base + lane_off;

    // One 16-column group: full K=16 reduction via 4 chained WMMAs.
    // B 4x16 f32 layout: VGPR0 = row (kbase+2*hi), VGPR1 = row+1;
    // N = lane&15 striped per lane half -> 64B/half-wave NT loads.
    auto group = [&](const float* gb) -> v8f {
      v8f c = {};
      v2f b;
      b.x = __builtin_nontemporal_load(gb);
      b.y = __builtin_nontemporal_load(gb + NN);
      c = __builtin_amdgcn_wmma_f32_16x16x4_f32(false, a0, false, b, (short)0,
                                                c, false, false);
      gb += 4 * NN;
      b.x = __builtin_nontemporal_load(gb);
      b.y = __builtin_nontemporal_load(gb + NN);
      c = __builtin_amdgcn_wmma_f32_16x16x4_f32(false, a1, false, b, (short)0,
                                                c, false, false);
      gb += 4 * NN;
      b.x = __builtin_nontemporal_load(gb);
      b.y = __builtin_nontemporal_load(gb + NN);
      c = __builtin_amdgcn_wmma_f32_16x16x4_f32(false, a2, false, b, (short)0,
                                                c, false, false);
      gb += 4 * NN;
      b.x = __builtin_nontemporal_load(gb);
      b.y = __builtin_nontemporal_load(gb + NN);
      c = __builtin_amdgcn_wmma_f32_16x16x4_f32(false, a3, false, b, (short)0,
                                                c, false, false);
      return c;
    };

    const v8f c0 = group(wbase);
    const v8f c1 = group(wbase + 16);
    const v8f c2 = group(wbase + 32);
    const v8f c3 = group(wbase + 48);

    // D rows identical; VGPR0 holds row0 (lanes 0-15) / row8 (lanes 16-31),
    // same values. Pack two column groups per store -> full 128B b32 NT stores.
    const float v01 = hi ? c1[0] : c0[0];
    const float v23 = hi ? c3[0] : c2[0];
    __builtin_nontemporal_store(v01, out + base + (size_t)lane);
    __builtin_nontemporal_store(v23, out + base + 32 + (size_t)lane);
  } else {
    // Scalar tail (never taken for 2048x2048, kept for safety; no WMMA here).
    for (size_t n = base + (size_t)lane; n < NN; n += 32) {
      float s;
      s = w0 * adj[n];
      s = fmaf(w1, adj[1 * NN + n], s);
      s = fmaf(w2, adj[2 * NN + n], s);
      s = fmaf(w3, adj[3 * NN + n], s);
      s = fmaf(w4, adj[4 * NN + n], s);
      s = fmaf(w5, adj[5 * NN + n], s);
      s = fmaf(w6, adj[6 * NN + n], s);
      s = fmaf(w7, adj[7 * NN + n], s);
      s = fmaf(w8, adj[8 * NN + n], s);
      s = fmaf(w9, adj[9 * NN + n], s);
      s = fmaf(w10, adj[10 * NN + n], s);
      s = fmaf(w11, adj[11 * NN + n], s);
      s = fmaf(w12, adj[12 * NN + n], s);
      s = fmaf(w13, adj[13 * NN + n], s);
      s = fmaf(w14, adj[14 * NN + n], s);
      s = fmaf(w15, adj[15 * NN + n], s);
      out[n] = s;
    }
  }
}

extern "C" void kernel_launch(void* const* d_in, const int* in_sizes, int n_in,
                              void* d_out, int out_size, void* d_ws, size_t ws_size,
                              hipStream_t stream) {
  const float* adj = (const float*)d_in[0];
  const float* lam = (const float*)d_in[1];
  float* out = (float*)d_out;

  const int NN = in_sizes[0] / TSTEPS;  // elements per slice == out_size
  const int blocks = (NN + 511) / 512;  // 512 outputs per 256-thread block
  ewma_fold_wmma<<<blocks, 256, 0, stream>>>(adj, lam, out, NN);
}